// NMP_Conv_69681549410767
// MI455X (gfx1250) — compile-verified
//
#include <hip/hip_runtime.h>
#include <hip/hip_bf16.h>

#define N_NODES   50000
#define N_EDGES   800000
#define F_IN      16
#define F_EDGE    8
#define H_DIM     8
#define NUM_GRAPHS 512

typedef __attribute__((ext_vector_type(2))) float v2f;
typedef __attribute__((ext_vector_type(8))) float v8f;

static_assert(N_NODES % 16 == 0, "WMMA node tiles assume N_NODES % 16 == 0");

// ---------------------------------------------------------------------------
// Zero a small float buffer (pooled accumulator).
__global__ void k_zero(float* __restrict__ p, int n) {
    int i = blockIdx.x * blockDim.x + threadIdx.x;
    if (i < n) p[i] = 0.f;
}

// ---------------------------------------------------------------------------
// Fused per-node precompute for one NNConv layer, all via WMMA f32 16x16x4.
// One wave owns a 16-node tile of A = f(Asrc) ([N x IN_DIM], f = optional ReLU)
// and computes an 80-column GEMM in 5 column tiles of 16:
//   cols  0..63 : T[n, j*8+o]   = sum_k A[n,k] * We[j, k*8+o]   (edge-weight basis)
//   cols 64..71 : agg[n, o]     = b[o] + sum_k A[n,k] * root[k*8+o]  (root/self term)
//   cols 72..79 : Bterm[n, o]   =        sum_k A[n,k] * be[k*8+o]    (edge-bias term)
//
// f32 16x4 A-operand layout (2 VGPRs): lane = (half<<4)|m -> row M=m,
// K = half*2 + {0,1}; B-operand symmetric; C/D: VGPR r -> M = half*8+r, N = m.
template <int IN_DIM, bool RELU>
__global__ void k_pre_wmma(const float* __restrict__ Asrc,   // [N, IN_DIM]
                           const float* __restrict__ We,     // [F_EDGE, IN_DIM*8]
                           const float* __restrict__ root,   // [IN_DIM, 8]
                           const float* __restrict__ bvec,   // [8]
                           const float* __restrict__ be,     // [IN_DIM*8]
                           float* __restrict__ T,            // [N, 64]
                           float* __restrict__ agg,          // [N, 8]
                           float* __restrict__ Bout) {       // [N, 8]
    constexpr int KT = IN_DIM / 4;                 // K-steps of 4
    const int lane = threadIdx.x & 31;
    const int wave = blockIdx.x * (blockDim.x >> 5) + (threadIdx.x >> 5);
    const int base = wave * 16;
    if (base >= N_NODES) return;                   // wave-uniform guard only

    const int m    = lane & 15;                    // A row / B column within tile
    const int kk   = (lane >> 4) * 2;              // K pair held by this half-wave
    const int node = base + m;
    const int mbase = (lane >> 4) * 8;             // C/D row base for this half-wave
    const int wrow  = IN_DIM * H_DIM;              // We row stride

    // Preload this lane's A fragments once (8B-aligned vector loads), reuse
    // across all 5 column tiles. No per-lane guards -> EXEC stays all-ones.
    v2f a[KT];
    const float* arow = Asrc + (size_t)node * IN_DIM;
#pragma unroll
    for (int kt = 0; kt < KT; ++kt) {
        v2f t = *(const v2f*)(arow + kt * 4 + kk);
        if (RELU) { t.x = fmaxf(t.x, 0.f); t.y = fmaxf(t.y, 0.f); }
        a[kt] = t;
    }

#pragma unroll
    for (int nt = 0; nt < 5; ++nt) {
        const float* bp;
        int o;
        if (nt < 4) {                              // T columns: basis from We
            const int ncol = nt * 16 + m;
            bp = We + (ncol >> 3) * wrow;
            o  = ncol & 7;
        } else {                                   // root-term (m<8) / be-term (m>=8)
            bp = (m < 8) ? root : be;
            o  = m & 7;
        }
        v8f c = {0.f, 0.f, 0.f, 0.f, 0.f, 0.f, 0.f, 0.f};
#pragma unroll
        for (int kt = 0; kt < KT; ++kt) {
            const int k0 = kt * 4 + kk;
            v2f b;
            b.x = bp[(k0)     * H_DIM + o];
            b.y = bp[(k0 + 1) * H_DIM + o];
            c = __builtin_amdgcn_wmma_f32_16x16x4_f32(
                    /*neg_a=*/false, a[kt], /*neg_b=*/false, b,
                    /*c_mod=*/(short)0, c, /*reuse_a=*/false, /*reuse_b=*/false);
        }
        if (nt < 4) {
#pragma unroll
            for (int r = 0; r < 8; ++r)
                T[(size_t)(base + mbase + r) * 64 + nt * 16 + m] = c[r];
        } else if (m < 8) {
            const float bias = bvec[o];
#pragma unroll
            for (int r = 0; r < 8; ++r)
                agg[(size_t)(base + mbase + r) * 8 + o] = c[r] + bias;
        } else {
#pragma unroll
            for (int r = 0; r < 8; ++r)
                Bout[(size_t)(base + mbase + r) * 8 + o] = c[r];
        }
    }
}

// ---------------------------------------------------------------------------
// Per-edge message + scatter: acc[o] = B[s,o] + sum_j ea[e,j]*T[s, j*8+o];
// atomically accumulate into agg[d, o]. T/B/agg are all L2-resident.
__global__ void k_edge(const int* __restrict__ src, const int* __restrict__ dst,
                       const float* __restrict__ ea,
                       const float* __restrict__ T,
                       const float* __restrict__ Bsrc,
                       float* __restrict__ agg, int n_edges) {
    int e = blockIdx.x * blockDim.x + threadIdx.x;
    if (e >= n_edges) return;
    const int s = src[e];
    const int d = dst[e];

    const float4* eap = (const float4*)(ea + (size_t)e * 8);
    float4 e0 = eap[0], e1 = eap[1];
    float w[8] = {e0.x, e0.y, e0.z, e0.w, e1.x, e1.y, e1.z, e1.w};

    const float4* Bp = (const float4*)(Bsrc + (size_t)s * 8);
    float4 b0 = Bp[0], b1 = Bp[1];
    float acc[8] = {b0.x, b0.y, b0.z, b0.w, b1.x, b1.y, b1.z, b1.w};

    const float4* Tp = (const float4*)(T + (size_t)s * 64);
#pragma unroll
    for (int j = 0; j < 8; ++j) {
        float4 t0 = Tp[j * 2], t1 = Tp[j * 2 + 1];
        float wj = w[j];
        acc[0] += wj * t0.x; acc[1] += wj * t0.y; acc[2] += wj * t0.z; acc[3] += wj * t0.w;
        acc[4] += wj * t1.x; acc[5] += wj * t1.y; acc[6] += wj * t1.z; acc[7] += wj * t1.w;
    }
    float* ap = agg + (size_t)d * 8;
#pragma unroll
    for (int o = 0; o < 8; ++o) atomicAdd(ap + o, acc[o]);
}

// ---------------------------------------------------------------------------
// Graph sum-pooling: pooled[batch[n], o] += relu(agg2[n, o])
__global__ void k_pool(const float* __restrict__ agg2, const int* __restrict__ batch,
                       float* __restrict__ pooled, int n_nodes) {
    int i = blockIdx.x * blockDim.x + threadIdx.x;
    if (i >= n_nodes * H_DIM) return;
    int n = i >> 3, o = i & 7;
    float v = fmaxf(agg2[i], 0.f);
    atomicAdd(&pooled[batch[n] * H_DIM + o], v);
}

// ---------------------------------------------------------------------------
// Readout: out[g] = blast + pooled[g,:] . Wlast
__global__ void k_final(const float* __restrict__ pooled, const float* __restrict__ Wlast,
                        const float* __restrict__ blast, float* __restrict__ out) {
    int g = blockIdx.x * blockDim.x + threadIdx.x;
    if (g >= NUM_GRAPHS) return;
    float acc = blast[0];
#pragma unroll
    for (int o = 0; o < H_DIM; ++o) acc += pooled[g * H_DIM + o] * Wlast[o];
    out[g] = acc;
}

// ---------------------------------------------------------------------------
extern "C" void kernel_launch(void* const* d_in, const int* in_sizes, int n_in,
                              void* d_out, int out_size, void* d_ws, size_t ws_size,
                              hipStream_t stream) {
    const float* x     = (const float*)d_in[0];
    const int*   ei    = (const int*)  d_in[1];
    const float* ea    = (const float*)d_in[2];
    const int*   batch = (const int*)  d_in[3];
    const float* We1   = (const float*)d_in[4];
    const float* be1   = (const float*)d_in[5];
    const float* root1 = (const float*)d_in[6];
    const float* b1    = (const float*)d_in[7];
    const float* We2   = (const float*)d_in[8];
    const float* be2   = (const float*)d_in[9];
    const float* root2 = (const float*)d_in[10];
    const float* b2    = (const float*)d_in[11];
    const float* Wlast = (const float*)d_in[12];
    const float* blast = (const float*)d_in[13];

    const int* src = ei;             // edge_index[0]
    const int* dst = ei + N_EDGES;   // edge_index[1]
    float* out = (float*)d_out;

    // workspace layout (T and Bterm reused across the two layers)
    float* ws     = (float*)d_ws;
    float* T      = ws;                            // N * 64
    float* Bterm  = T     + (size_t)N_NODES * 64;  // N * 8
    float* agg1   = Bterm + (size_t)N_NODES * 8;   // N * 8
    float* agg2   = agg1  + (size_t)N_NODES * 8;   // N * 8
    float* pooled = agg2  + (size_t)N_NODES * 8;   // 512 * 8

    const int TPB = 256;
    const int g_no   = (N_NODES * H_DIM + TPB - 1) / TPB;
    const int g_edge = (N_EDGES + TPB - 1) / TPB;
    const int waves  = N_NODES / 16;                       // exact (50000/16 = 3125)
    const int g_wmma = (waves + (TPB / 32) - 1) / (TPB / 32);

    // pooled accumulator must start at zero every call
    k_zero<<<(NUM_GRAPHS * H_DIM + TPB - 1) / TPB, TPB, 0, stream>>>(pooled, NUM_GRAPHS * H_DIM);

    // ---- layer 1: T1 + agg1-init + B1 in one WMMA kernel, then edge scatter
    k_pre_wmma<F_IN, false><<<g_wmma, TPB, 0, stream>>>(x, We1, root1, b1, be1,
                                                        T, agg1, Bterm);
    k_edge<<<g_edge, TPB, 0, stream>>>(src, dst, ea, T, Bterm, agg1, N_EDGES);

    // ---- layer 2: ReLU(agg1) fused into A-fragment reads
    k_pre_wmma<H_DIM, true><<<g_wmma, TPB, 0, stream>>>(agg1, We2, root2, b2, be2,
                                                        T, agg2, Bterm);
    k_edge<<<g_edge, TPB, 0, stream>>>(src, dst, ea, T, Bterm, agg2, N_EDGES);

    // ---- pooling + readout ----
    k_pool<<<g_no, TPB, 0, stream>>>(agg2, batch, pooled, N_NODES);
    k_final<<<(NUM_GRAPHS + TPB - 1) / TPB, TPB, 0, stream>>>(pooled, Wlast, blast, out);
}